// SCLayer_44890998178478
// MI455X (gfx1250) — compile-verified
//
#include <hip/hip_runtime.h>
#include <hip/hip_bf16.h>

typedef float v2f __attribute__((ext_vector_type(2)));
typedef float v8f __attribute__((ext_vector_type(8)));

#define D128 128

// ---------------------------------------------------------------------------
// CDNA5 async global->LDS copy (ASYNCcnt path), per CDNA5 ISA
// ---------------------------------------------------------------------------
__device__ __forceinline__ void async_ld_b128(unsigned lds_off, const void* gaddr) {
    asm volatile("global_load_async_to_lds_b128 %0, %1, off"
                 :: "v"(lds_off), "v"(gaddr) : "memory");
}
__device__ __forceinline__ void wait_async0() {
    asm volatile("s_wait_asynccnt 0x0" ::: "memory");
}
__device__ __forceinline__ unsigned lds_off_of(const void* p) {
    return (unsigned)(size_t)p;    // LDS flat addr: low 32 bits = LDS offset
}

// ---------------------------------------------------------------------------
// zero fill
// ---------------------------------------------------------------------------
__global__ void k_zero(float* __restrict__ p, long n) {
    long i = (long)blockIdx.x * blockDim.x + threadIdx.x;
    long stride = (long)gridDim.x * blockDim.x;
    for (; i < n; i += stride) p[i] = 0.0f;
}

// ---------------------------------------------------------------------------
// edge scatter: agg[dst] += w * x[src]   (32 lanes per edge, float4 per lane)
// ---------------------------------------------------------------------------
__global__ void k_edge_scatter(const float* __restrict__ x,
                               const int* __restrict__ src,
                               const int* __restrict__ dst,
                               const float* __restrict__ w,
                               float* __restrict__ agg, int E) {
    int lane = threadIdx.x & 31;
    int eid = blockIdx.x * 8 + (threadIdx.x >> 5);
    if (eid >= E) return;
    int s = src[eid], d = dst[eid];
    float we = w[eid];
    const float4* xr = (const float4*)(x + (size_t)s * D128);
    float* ar = agg + (size_t)d * D128 + lane * 4;
    float4 v = xr[lane];
    atomicAdd(ar + 0, we * v.x);
    atomicAdd(ar + 1, we * v.y);
    atomicAdd(ar + 2, we * v.z);
    atomicAdd(ar + 3, we * v.w);
}

// ---------------------------------------------------------------------------
// WMMA GEMM:  C = act(A1 @ B1 [+ A2 @ B2] + bias), shapes [*,128]@[128,128].
// Block = 8 waves = 16 rows x 128 cols. A strips + 32-row B chunks staged in
// LDS via async loads; hot loop is ds_load + v_wmma_f32_16x16x4_f32.
// ---------------------------------------------------------------------------
template <int DUAL>
__global__ void k_gemm(const float* __restrict__ A1, const float* __restrict__ B1,
                       const float* __restrict__ A2, const float* __restrict__ B2,
                       const float* __restrict__ bias, float* __restrict__ C,
                       int nrows, int relu) {
    __shared__ __align__(16) float sA[DUAL + 1][16 * D128];  // 8 KB each
    __shared__ __align__(16) float sB[DUAL + 1][32 * D128];  // 16 KB each

    const int tid  = threadIdx.x;
    const int wave = tid >> 5, lane = tid & 31;
    const int half = lane >> 4, l = lane & 15;
    const int row0 = blockIdx.x * 16;
    const int col0 = wave * 16;

    // stage A strip(s): 16x128 floats = 512 float4, 2 per thread
    for (int i = tid; i < 16 * D128 / 4; i += 256) {
        int e = i * 4;
        int r = e >> 7, cc = e & 127;
        int ra = row0 + r; if (ra >= nrows) ra = nrows - 1;
        async_ld_b128(lds_off_of(&sA[0][e]), A1 + (size_t)ra * D128 + cc);
        if (DUAL)
            async_ld_b128(lds_off_of(&sA[1][e]), A2 + (size_t)ra * D128 + cc);
    }

    v8f c = {};
    for (int kc = 0; kc < D128; kc += 32) {
        // stage B chunk [kc, kc+32) x 128 = 1024 float4, 4 per thread
        for (int i = tid; i < 32 * D128 / 4; i += 256) {
            int e = i * 4;
            async_ld_b128(lds_off_of(&sB[0][e]), B1 + (size_t)kc * D128 + e);
            if (DUAL)
                async_ld_b128(lds_off_of(&sB[1][e]), B2 + (size_t)kc * D128 + e);
        }
        if (kc + 32 < D128) __builtin_prefetch(B1 + (size_t)(kc + 32) * D128, 0, 1);
        wait_async0();
        __syncthreads();

        #pragma unroll
        for (int kk = 0; kk < 32; kk += 4) {
            int k0 = kk + 2 * half;              // A layout: lanes 0-15 K={0,1}, 16-31 K={2,3}
            v2f a1; a1.x = sA[0][l * D128 + kc + k0];
                    a1.y = sA[0][l * D128 + kc + k0 + 1];
            v2f b1; b1.x = sB[0][k0 * D128 + col0 + l];
                    b1.y = sB[0][(k0 + 1) * D128 + col0 + l];
            c = __builtin_amdgcn_wmma_f32_16x16x4_f32(false, a1, false, b1,
                                                      (short)0, c, false, false);
            if (DUAL) {
                v2f a2; a2.x = sA[1][l * D128 + kc + k0];
                        a2.y = sA[1][l * D128 + kc + k0 + 1];
                v2f b2; b2.x = sB[1][k0 * D128 + col0 + l];
                        b2.y = sB[1][(k0 + 1) * D128 + col0 + l];
                c = __builtin_amdgcn_wmma_f32_16x16x4_f32(false, a2, false, b2,
                                                          (short)0, c, false, false);
            }
        }
        __syncthreads();
    }

    float bcol = bias ? bias[col0 + l] : 0.0f;
    #pragma unroll
    for (int r = 0; r < 8; r++) {
        int row = row0 + r + 8 * half;           // C layout: VGPR r -> M=r / M=r+8
        if (row < nrows) {
            float v = c[r] + bcol;
            if (relu) v = fmaxf(v, 0.0f);
            C[(size_t)row * D128 + col0 + l] = v;
        }
    }
}

// ---------------------------------------------------------------------------
// row softmax over K=128: one wave per row (4 floats per lane)
// ---------------------------------------------------------------------------
__global__ void k_softmax(const float* __restrict__ s, float* __restrict__ probs, int nrows) {
    int lane = threadIdx.x & 31;
    int row = blockIdx.x * 8 + (threadIdx.x >> 5);
    if (row >= nrows) return;
    float4 v = ((const float4*)(s + (size_t)row * D128))[lane];
    float m = fmaxf(fmaxf(v.x, v.y), fmaxf(v.z, v.w));
    for (int off = 16; off; off >>= 1) m = fmaxf(m, __shfl_xor(m, off, 32));
    v.x = __expf(v.x - m); v.y = __expf(v.y - m);
    v.z = __expf(v.z - m); v.w = __expf(v.w - m);
    float sum = v.x + v.y + v.z + v.w;
    for (int off = 16; off; off >>= 1) sum += __shfl_xor(sum, off, 32);
    float inv = 1.0f / sum;
    v.x *= inv; v.y *= inv; v.z *= inv; v.w *= inv;
    ((float4*)(probs + (size_t)row * D128))[lane] = v;
}

// ---------------------------------------------------------------------------
// deg[i] = sum over edges with src==i of w
// ---------------------------------------------------------------------------
__global__ void k_deg(const int* __restrict__ src, const float* __restrict__ w,
                      float* __restrict__ deg, int E) {
    int e = blockIdx.x * blockDim.x + threadIdx.x;
    if (e < E) atomicAdd(&deg[src[e]], w[e]);
}

// ---------------------------------------------------------------------------
// mincut_num = sum_e w_e * <probs[src_e], probs[dst_e]>   (wave per edge)
// ---------------------------------------------------------------------------
__global__ void k_edge_dot(const float* __restrict__ probs,
                           const int* __restrict__ src, const int* __restrict__ dst,
                           const float* __restrict__ w, float* __restrict__ out_num, int E) {
    __shared__ float part;
    if (threadIdx.x == 0) part = 0.0f;
    __syncthreads();
    int lane = threadIdx.x & 31;
    int eid = blockIdx.x * 8 + (threadIdx.x >> 5);
    float acc = 0.0f;
    if (eid < E) {
        const float4* ps = (const float4*)(probs + (size_t)src[eid] * D128);
        const float4* pd = (const float4*)(probs + (size_t)dst[eid] * D128);
        float4 a = ps[lane], b = pd[lane];
        acc = a.x * b.x + a.y * b.y + a.z * b.z + a.w * b.w;
    }
    for (int off = 16; off; off >>= 1) acc += __shfl_xor(acc, off, 32);
    if (lane == 0 && eid < E) atomicAdd(&part, acc * w[eid]);   // ds_add_f32
    __syncthreads();
    if (threadIdx.x == 0) atomicAdd(out_num, part);
}

// ---------------------------------------------------------------------------
// mincut_den = sum_i deg_i * ||probs_i||^2  (wave per node)
// ---------------------------------------------------------------------------
__global__ void k_den(const float* __restrict__ probs, const float* __restrict__ deg,
                      float* __restrict__ out_den, int nrows) {
    __shared__ float part;
    if (threadIdx.x == 0) part = 0.0f;
    __syncthreads();
    int lane = threadIdx.x & 31;
    int row = blockIdx.x * 8 + (threadIdx.x >> 5);
    float acc = 0.0f;
    if (row < nrows) {
        float4 v = ((const float4*)(probs + (size_t)row * D128))[lane];
        acc = v.x * v.x + v.y * v.y + v.z * v.z + v.w * v.w;
    }
    for (int off = 16; off; off >>= 1) acc += __shfl_xor(acc, off, 32);
    if (lane == 0 && row < nrows) atomicAdd(&part, acc * deg[row]);
    __syncthreads();
    if (threadIdx.x == 0) atomicAdd(out_den, part);
}

__global__ void k_mc_final(const float* __restrict__ scal, float* __restrict__ mc) {
    mc[0] = -(scal[0] / scal[1]);
}

// ---------------------------------------------------------------------------
// ss = probs^T @ probs. Each block stages 32-row chunks of probs in LDS and
// computes ALL 64 16x16 tiles from it (wave w -> row-tile w, 8 col tiles,
// 8 accumulators, 64 WMMA per chunk). Partials atomicAdd'ed into ss.
// ---------------------------------------------------------------------------
__global__ void k_gram_wmma(const float* __restrict__ probs, float* __restrict__ ss, int nrows) {
    __shared__ __align__(16) float sP[32 * D128];   // 16 KB
    const int tid  = threadIdx.x;
    const int wave = tid >> 5, lane = tid & 31;
    const int half = lane >> 4, l = lane & 15;
    const int m0 = wave * 16;

    v8f acc[8];
    #pragma unroll
    for (int t = 0; t < 8; t++) { v8f z = {}; acc[t] = z; }

    for (int base = blockIdx.x * 32; base < nrows; base += gridDim.x * 32) {
        int nrem = nrows - base; if (nrem > 32) nrem = 32;
        for (int i = tid; i < 32 * D128 / 4; i += 256) {
            int e = i * 4;
            int r = e >> 7;
            if (r < nrem) {
                async_ld_b128(lds_off_of(&sP[e]), probs + (size_t)(base + r) * D128 + (e & 127));
            } else {
                float4 z = {0.f, 0.f, 0.f, 0.f};
                *(float4*)&sP[e] = z;            // zero-pad tail rows
            }
        }
        wait_async0();
        __syncthreads();

        #pragma unroll
        for (int kk = 0; kk < 32; kk += 4) {
            int k0 = kk + 2 * half;
            v2f a; a.x = sP[k0 * D128 + m0 + l];
                   a.y = sP[(k0 + 1) * D128 + m0 + l];
            #pragma unroll
            for (int t = 0; t < 8; t++) {
                v2f b; b.x = sP[k0 * D128 + t * 16 + l];
                       b.y = sP[(k0 + 1) * D128 + t * 16 + l];
                acc[t] = __builtin_amdgcn_wmma_f32_16x16x4_f32(false, a, false, b,
                                                               (short)0, acc[t], false, false);
            }
        }
        __syncthreads();
    }

    #pragma unroll
    for (int t = 0; t < 8; t++)
        #pragma unroll
        for (int r = 0; r < 8; r++)
            atomicAdd(&ss[(size_t)(m0 + r + 8 * half) * D128 + t * 16 + l], acc[t][r]);
}

// ---------------------------------------------------------------------------
// o_loss = || ss/||ss||_F - I/sqrt(K) ||_F   (single block, two phases)
// ---------------------------------------------------------------------------
__global__ void k_oloss(const float* __restrict__ ss, float* __restrict__ out) {
    __shared__ float red[256];
    int tid = threadIdx.x;
    float acc = 0.0f;
    for (int i = tid; i < D128 * D128; i += 256) { float v = ss[i]; acc += v * v; }
    red[tid] = acc; __syncthreads();
    for (int s = 128; s > 0; s >>= 1) { if (tid < s) red[tid] += red[tid + s]; __syncthreads(); }
    float inv = 1.0f / sqrtf(red[0]);
    __syncthreads();
    const float diag = 0.08838834764831845f;  // 1/sqrt(128)
    acc = 0.0f;
    for (int i = tid; i < D128 * D128; i += 256) {
        float v = ss[i] * inv - (((i % (D128 + 1)) == 0) ? diag : 0.0f);
        acc += v * v;
    }
    red[tid] = acc; __syncthreads();
    for (int s = 128; s > 0; s >>= 1) { if (tid < s) red[tid] += red[tid + s]; __syncthreads(); }
    if (tid == 0) out[0] = sqrtf(red[0]);
}

// ---------------------------------------------------------------------------
extern "C" void kernel_launch(void* const* d_in, const int* in_sizes, int n_in,
                              void* d_out, int out_size, void* d_ws, size_t ws_size,
                              hipStream_t stream) {
    const float* x     = (const float*)d_in[0];
    const int*   ei    = (const int*)d_in[1];
    const float* ew    = (const float*)d_in[2];
    const float* Wrel1 = (const float*)d_in[3];
    const float* brel1 = (const float*)d_in[4];
    const float* Wroot1= (const float*)d_in[5];
    const float* Wrel2 = (const float*)d_in[6];
    const float* brel2 = (const float*)d_in[7];
    const float* Wroot2= (const float*)d_in[8];
    const float* Wm1   = (const float*)d_in[9];
    const float* bm1   = (const float*)d_in[10];
    const float* Wout  = (const float*)d_in[11];
    const float* bout  = (const float*)d_in[12];

    const long NK = in_sizes[0];          // N * 128
    const int  Nn = (int)(NK / D128);     // 50000
    const int  E  = in_sizes[1] / 2;      // 800000
    const int* src = ei;
    const int* dst = ei + E;

    float* out   = (float*)d_out;
    float* probs = out;                   // [N,128]
    float* mc    = out + NK;              // scalar
    float* ol    = mc + 1;                // scalar

    float* f0   = (float*)d_ws;           // agg / logits reuse
    float* f1   = f0 + NK;                // h1
    float* f2   = f1 + NK;                // h2
    float* f3   = f2 + NK;                // mlp hidden
    float* deg  = f3 + NK;                // [N]
    float* ss   = deg + Nn;               // [128,128]
    float* scal = ss + D128 * D128;       // num, den

    const int rowBlocks = (Nn + 15) / 16;
    const int edgeBlk8  = (E + 7) / 8;
    const int nodeBlk8  = (Nn + 7) / 8;

    // zero scratch accumulators (deg/ss/scal are contiguous after f3)
    k_zero<<<2048, 256, 0, stream>>>(f0, NK);
    k_zero<<<256, 256, 0, stream>>>(deg, (long)Nn + D128 * D128 + 8);

    // ---- GraphConv layer 1 ----
    k_edge_scatter<<<edgeBlk8, 256, 0, stream>>>(x, src, dst, ew, f0, E);
    k_gemm<1><<<rowBlocks, 256, 0, stream>>>(f0, Wrel1, x, Wroot1, brel1, f1, Nn, 1);

    // ---- GraphConv layer 2 ----
    k_zero<<<2048, 256, 0, stream>>>(f0, NK);
    k_edge_scatter<<<edgeBlk8, 256, 0, stream>>>(f1, src, dst, ew, f0, E);
    k_gemm<1><<<rowBlocks, 256, 0, stream>>>(f0, Wrel2, f1, Wroot2, brel2, f2, Nn, 1);

    // ---- MLP + logits ----
    k_gemm<0><<<rowBlocks, 256, 0, stream>>>(f2, Wm1, nullptr, nullptr, bm1, f3, Nn, 0);
    k_gemm<0><<<rowBlocks, 256, 0, stream>>>(f3, Wout, nullptr, nullptr, bout, f0, Nn, 0);

    // ---- softmax -> probs (d_out) ----
    k_softmax<<<nodeBlk8, 256, 0, stream>>>(f0, probs, Nn);

    // ---- mincut loss ----
    k_deg<<<(E + 255) / 256, 256, 0, stream>>>(src, ew, deg, E);
    k_edge_dot<<<edgeBlk8, 256, 0, stream>>>(probs, src, dst, ew, &scal[0], E);
    k_den<<<nodeBlk8, 256, 0, stream>>>(probs, deg, &scal[1], Nn);
    k_mc_final<<<1, 1, 0, stream>>>(scal, mc);

    // ---- orthogonality loss ----
    k_gram_wmma<<<256, 256, 0, stream>>>(probs, ss, Nn);
    k_oloss<<<1, 256, 0, stream>>>(ss, ol);
}